// PatchPermAugmentation_36618891166040
// MI455X (gfx1250) — compile-verified
//
#include <hip/hip_runtime.h>

// PatchPermAugmentation: out[b,t,n,0] = x[b, 2*perm[b,t/2] + (t&1), n, 0]
//                        out[b,t,n,1:3] = x[b,t,n,1:3]
// Pure data movement, HBM-bound (~402 MB total => ~17us @ 23.3 TB/s).
// Strategy: wave32-coalesced gather/copy; per-block-uniform perm fetch via
// scalar load; non-temporal stores so the 192 MiB input stays resident in the
// 192 MB L2 and the duplicated channel-0 gather reads hit cache, not HBM.

namespace {
constexpr unsigned Bc  = 32;    // batch
constexpr unsigned Tc  = 1024;  // time
constexpr unsigned Nc  = 512;   // series
constexpr unsigned Cc  = 3;     // channels
constexpr unsigned PS  = 2;     // patch size
constexpr unsigned NB  = Tc / PS;        // 512 blocks per batch
constexpr unsigned THREADS = 256;        // 8 waves (wave32) per workgroup
// one thread per (b, block, n); each handles both t-halves of the patch
constexpr unsigned TOTAL_THREADS = Bc * NB * Nc;          // 8,388,608
constexpr unsigned GRID = TOTAL_THREADS / THREADS;        // 32,768 blocks
}  // namespace

__global__ __launch_bounds__(THREADS) void patch_perm_kernel(
    const float* __restrict__ x,      // (B, T, N, C) f32
    const int*   __restrict__ perm,   // (B, NB) i32
    float*       __restrict__ out)    // (B, T, N, C) f32
{
    // blockDim=256, N=512  =>  2 blocks per (b, blk) patch-row.
    // r = b*NB + blk is uniform across the whole workgroup.
    const unsigned r   = blockIdx.x >> 1;            // b*NB + blk (uniform)
    const unsigned b   = r >> 9;                     // / NB
    const unsigned blk = r & (NB - 1u);
    const unsigned n   = ((blockIdx.x & 1u) << 8) | threadIdx.x;

    // Uniform address -> compiler emits a scalar (SMEM) load, one per wave.
    const int p = perm[r];

    const size_t row = (size_t)Nc * Cc;              // floats per t-row (1536)
    const size_t dst = ((size_t)(b * Tc + PS * blk) * Nc + n) * Cc;
    const size_t src = ((size_t)(b * Tc + PS * (unsigned)p) * Nc + n) * Cc;

    // Pass-through channels 1..2 for both halves of the patch (regular loads:
    // these lines also feed other threads' gathers -> keep them temporal).
    const float a1 = x[dst + 1];
    const float a2 = x[dst + 2];
    const float c1 = x[dst + row + 1];
    const float c2 = x[dst + row + 2];
    // Gathered channel 0 for both halves (hits L2 thanks to NT stores below).
    const float g0 = x[src];
    const float g1 = x[src + row];

    // Write-once output stream: non-temporal stores (TH_STORE_NT) so the
    // store stream does not evict the input working set from L2.
    float* o0 = out + dst;
    __builtin_nontemporal_store(g0, o0 + 0);
    __builtin_nontemporal_store(a1, o0 + 1);
    __builtin_nontemporal_store(a2, o0 + 2);
    float* o1 = o0 + row;
    __builtin_nontemporal_store(g1, o1 + 0);
    __builtin_nontemporal_store(c1, o1 + 1);
    __builtin_nontemporal_store(c2, o1 + 2);
}

extern "C" void kernel_launch(void* const* d_in, const int* in_sizes, int n_in,
                              void* d_out, int out_size, void* d_ws, size_t ws_size,
                              hipStream_t stream) {
    (void)in_sizes; (void)n_in; (void)out_size; (void)d_ws; (void)ws_size;
    const float* x    = (const float*)d_in[0];   // (B,T,N,C) f32
    const int*   perm = (const int*)d_in[1];     // (B,NB)    i32
    float*       out  = (float*)d_out;           // (B,T,N,C) f32

    patch_perm_kernel<<<dim3(GRID), dim3(THREADS), 0, stream>>>(x, perm, out);
}